// CubicSpline_14156212207867
// MI455X (gfx1250) — compile-verified
//
#include <hip/hip_runtime.h>
#include <stdint.h>

// ---------------------------------------------------------------------------
// Uniform-knot cubic Hermite spline (65 knots on [-2,2], h = 2^-4) with linear
// extrapolation, evaluated over 67.1M fp32 elements.
//
// Roofline: 537 MB touch-once traffic -> ~23 us at 23.3 TB/s.  No matmul
// structure -> WMMA inapplicable.  CDNA5 paths used instead:
//   * TDM tensor_load_to_lds stages knot tables into LDS (s_wait_tensorcnt)
//   * 66-entry coefficient table in LDS, one ds_load_b128 per element;
//     entries 0 and 65 are degenerate-linear segments so extrapolation is
//     handled by the same select-free Horner path (i = med3(floor(s),-1,64))
//   * non-temporal b128 global loads/stores (don't thrash the 192 MB L2)
//   * 2 x float4 per thread to amortize per-thread overhead (wave32 blocks)
// ---------------------------------------------------------------------------

typedef float        v4f   __attribute__((ext_vector_type(4)));
typedef unsigned int u32x4 __attribute__((ext_vector_type(4)));
typedef int          i32x4 __attribute__((ext_vector_type(4)));
typedef int          i32x8 __attribute__((ext_vector_type(8)));

#define NK   65
#define NSEG 64
#define XL   (-2.0f)
#define HSEG (0.0625f)   /* 4/64 = 2^-4, exact in fp32 */
#define INVH (16.0f)

#if __has_builtin(__builtin_amdgcn_tensor_load_to_lds) && \
    __has_builtin(__builtin_amdgcn_s_wait_tensorcnt)
#define USE_TDM 1
#else
#define USE_TDM 0
#endif

#if USE_TDM
// D# for a 1-D row of `nelem` 4-byte elements (global -> LDS), per CDNA5 ISA
// CDNA5 ISA.  Groups 2/3 zero (<=2D tensor, no cluster).
__device__ __forceinline__ void tdm_load_row_f32(uint32_t lds_byte_addr,
                                                 const void* gptr,
                                                 uint32_t nelem) {
  uint64_t ga = (uint64_t)(uintptr_t)gptr;
  u32x4 g0;
  g0[0] = 1u;                                                // count=1 (valid)
  g0[1] = lds_byte_addr;                                     // lds_addr
  g0[2] = (uint32_t)(ga & 0xFFFFFFFFu);                      // global_addr lo
  g0[3] = (uint32_t)((ga >> 32) & 0x01FFFFFFu) | (2u << 30); // addr hi | type=2

  i32x8 g1;
  g1[0] = (int)(2u << 16);                 // workgroup_mask=0, data_size=4B
  g1[1] = (int)((nelem & 0xFFFFu) << 16);  // tensor_dim0[15:0] @ bits 63:48
  g1[2] = (int)(1u << 16);                 // tensor_dim0 hi=0, tensor_dim1=1
  g1[3] = (int)((nelem & 0xFFFFu) << 16);  // tensor_dim1 hi=0, tile_dim0=nelem
  g1[4] = 1;                               // tile_dim1=1, tile_dim2=0
  g1[5] = (int)nelem;                      // tensor_dim0_stride[31:0]
  g1[6] = 0;
  g1[7] = 0;

  i32x4 z4 = {0, 0, 0, 0};
#if defined(__clang_major__) && __clang_major__ >= 23
  i32x8 z8 = {0, 0, 0, 0, 0, 0, 0, 0};
  __builtin_amdgcn_tensor_load_to_lds(g0, g1, z4, z4, z8, 0);
#else
  __builtin_amdgcn_tensor_load_to_lds(g0, g1, z4, z4, 0);
#endif
}
#endif  // USE_TDM

// coef[j] = (y0, s0, b, a); j = i+1 where i in [-1, 64].  Entries 0 and 65
// are degenerate-linear (a=b=0) so extrapolation uses the same Horner.
__device__ __forceinline__ float eval_spline(float t,
                                             const v4f* __restrict__ coef) {
  float s = __builtin_fmaf(t, INVH, 32.0f);    // (t - XL) * 16
  float f = floorf(s);
  int   i = (int)f;
  i = i < -1 ? -1 : (i > NSEG ? NSEG : i);     // v_med3_i32(i, -1, 64)
  float u = s - (float)i;                      // exact frac(s) when in range
  v4f   c = coef[i + 1];                       // ds_load_b128
  return __builtin_fmaf(
             __builtin_fmaf(__builtin_fmaf(c.w, u, c.z), u, c.y), u, c.x);
}

__global__ void spline_eval_kernel(const float* __restrict__ t,
                                   const float* __restrict__ y,
                                   const float* __restrict__ ys,
                                   const float* __restrict__ y1p,
                                   const float* __restrict__ y2p,
                                   float* __restrict__ out,
                                   long long n) {
  __shared__ float sy[NK];
  __shared__ float sys[NK];
  __shared__ v4f   coef[NK + 1];   // 66 entries, 1056 B

#if USE_TDM
  // Wave 0 issues both TDM descriptors; TDM ignores EXEC, so the divergent
  // guard only makes the other waves skip via EXECZ branch.
  if (threadIdx.x == 0) {
    uint32_t lds_y  = (uint32_t)(size_t)(__attribute__((address_space(3))) float*)sy;
    uint32_t lds_ys = (uint32_t)(size_t)(__attribute__((address_space(3))) float*)sys;
    tdm_load_row_f32(lds_y,  y,  NK);
    tdm_load_row_f32(lds_ys, ys, NK);
    __builtin_amdgcn_s_wait_tensorcnt(0);
  }
#else
  for (int i = (int)threadIdx.x; i < NK; i += (int)blockDim.x) {
    sy[i]  = y[i];
    sys[i] = ys[i];
  }
#endif
  __syncthreads();

  float y1v = y1p[0];   // uniform scalar loads (f(-2), f(+2))
  float y2v = y2p[0];

  // Build the 66-entry coefficient table.
  if (threadIdx.x < NSEG) {
    int   i  = (int)threadIdx.x;
    float y0 = sy[i],  yn = sy[i + 1];
    float s0 = sys[i], s1 = sys[i + 1];
    v4f c;
    c.x = y0;
    c.y = s0;
    c.z = __builtin_fmaf(-3.0f, y0, 3.0f * yn) - 2.0f * s0 - s1;  // u^2
    c.w = __builtin_fmaf(2.0f, y0, -2.0f * yn) + s0 + s1;         // u^3
    coef[i + 1] = c;
  } else if (threadIdx.x == NSEG) {          // left-linear: y1 + (t - XL)
    v4f c; c.x = y1v - HSEG; c.y = HSEG; c.z = 0.0f; c.w = 0.0f;
    coef[0] = c;
  } else if (threadIdx.x == NSEG + 1) {      // right-linear: y2 + (t - XR)
    v4f c; c.x = y2v; c.y = HSEG; c.z = 0.0f; c.w = 0.0f;
    coef[NK] = c;
  }
  __syncthreads();

  // 2 x float4 per thread, coalesced: block covers 2*blockDim vec4s.
  long long nvec = n >> 2;
  long long v0   = (long long)blockIdx.x * (blockDim.x * 2) + threadIdx.x;
  long long v1   = v0 + blockDim.x;

  const v4f* tp = (const v4f*)t;
  v4f*       op = (v4f*)out;

  if (v0 < nvec) {
    v4f tv0 = __builtin_nontemporal_load(tp + v0);
    bool has1 = (v1 < nvec);
    v4f tv1;
    if (has1) tv1 = __builtin_nontemporal_load(tp + v1);

    v4f r0;
    r0.x = eval_spline(tv0.x, coef);
    r0.y = eval_spline(tv0.y, coef);
    r0.z = eval_spline(tv0.z, coef);
    r0.w = eval_spline(tv0.w, coef);
    __builtin_nontemporal_store(r0, op + v0);

    if (has1) {
      v4f r1;
      r1.x = eval_spline(tv1.x, coef);
      r1.y = eval_spline(tv1.y, coef);
      r1.z = eval_spline(tv1.z, coef);
      r1.w = eval_spline(tv1.w, coef);
      __builtin_nontemporal_store(r1, op + v1);
    }
  }

  // Scalar tail, n % 4 elements (not hit for the 8x4096x2048 shape).
  if (blockIdx.x == 0 && threadIdx.x < (unsigned)(n & 3)) {
    long long e = (nvec << 2) + threadIdx.x;
    out[e] = eval_spline(t[e], coef);
  }
}

extern "C" void kernel_launch(void* const* d_in, const int* in_sizes, int n_in,
                              void* d_out, int out_size, void* d_ws, size_t ws_size,
                              hipStream_t stream) {
  (void)n_in; (void)d_ws; (void)ws_size; (void)out_size;
  const float* t   = (const float*)d_in[0];
  // d_in[1] = x_knots: uniform linspace(-2,2,65); replaced by index math.
  const float* y   = (const float*)d_in[2];
  const float* ys  = (const float*)d_in[3];
  const float* y1p = (const float*)d_in[4];
  const float* y2p = (const float*)d_in[5];
  float* out = (float*)d_out;

  long long n    = (long long)in_sizes[0];       // 67,108,864
  long long nvec = n >> 2;                       // float4 count
  const int block = 256;                         // 8 wave32 waves
  long long vec_per_block = (long long)block * 2;
  long long grid = (nvec + vec_per_block - 1) / vec_per_block;
  if (grid < 1) grid = 1;                        // still run tail path

  spline_eval_kernel<<<(unsigned int)grid, block, 0, stream>>>(
      t, y, ys, y1p, y2p, out, n);
}